// OrthogonalBlock_1752346657580
// MI455X (gfx1250) — compile-verified
//
#include <hip/hip_runtime.h>
#include <hip/hip_bf16.h>

// CholeskyQR implementation of column-wise Gram-Schmidt:
//   G = X^T X ; G = R^T R (Cholesky) ; Q = X R^{-1}
// Mathematically identical Q to modified Gram-Schmidt (positive diag R).
// Bandwidth-optimal on MI455X: 2 reads + 1 write of the 128MB matrix instead
// of ~64 passes for the sequential MGS loop. fp32 WMMA (V_WMMA_F32_16X16X4_F32)
// for both GEMMs; Tensor Data Mover (TDM) double-buffered global->LDS staging.

typedef __attribute__((ext_vector_type(2))) float v2f;
typedef __attribute__((ext_vector_type(8))) float v8f;
typedef __attribute__((ext_vector_type(4))) unsigned int v4u;
typedef __attribute__((ext_vector_type(8))) int v8i;
typedef __attribute__((ext_vector_type(4))) int v4i;

#define LDSS 72   // padded LDS row stride (floats): 64 data + 8 pad (bank spread)
#define MCOLS 64

#if defined(__AMDGCN__) && __has_builtin(__builtin_amdgcn_tensor_load_to_lds) && \
    __has_builtin(__builtin_amdgcn_s_wait_tensorcnt)
#define USE_TDM 1
#else
#define USE_TDM 0
#endif

__global__ void zero_kernel(float* __restrict__ p, int n) {
    int i = blockIdx.x * blockDim.x + threadIdx.x;
    if (i < n) p[i] = 0.0f;
}

#if USE_TDM
// ---------------------------------------------------------------------------
// TDM: DMA a 64-row x 64-col fp32 tile from global to LDS.
//  - LDS rows padded to 72 floats via D# pad (interval=64 dwords -> code 5,
//    amount=8 dwords -> code 7).
//  - tensor_dim1 = rows remaining in matrix -> OOB rows read as zero (tail).
// Issued by one wave; completion via TENSORcnt.
// This toolchain exposes the 6-arg builtin:
//   (uint32x4 g0, int32x8 g1, int32x4 g2, int32x4 g3, int32x8 g4, i32 cpol)
// 2D tensor: groups 2..4 are zero.
// ---------------------------------------------------------------------------
__device__ __forceinline__ void tdm_load_tile(const float* gsrc,
                                              unsigned lds_byte_off,
                                              unsigned rows_remaining) {
    unsigned long long ga = (unsigned long long)(uintptr_t)gsrc;

    v4u g0;
    g0[0] = 1u;                                   // count=1 (valid), user mode
    g0[1] = lds_byte_off;                         // lds_addr (bytes)
    g0[2] = (unsigned)(ga & 0xFFFFFFFFu);         // global_addr[31:0]
    g0[3] = (unsigned)((ga >> 32) & 0x01FFFFFFu)  // global_addr[56:32]
            | (2u << 30);                         // type=2 ("image")

    v8i g1;
    g1[0] = (int)((2u << 16)      // data_size = 4 bytes
                | (1u << 20)      // pad_enable
                | (5u << 22)      // pad_interval: 64 dwords
                | (7u << 25));    // pad_amount: 8 dwords
    g1[1] = (int)(64u << 16);                                  // tensor_dim0 = 64
    g1[2] = (int)((rows_remaining & 0xFFFFu) << 16);           // tensor_dim1 lo
    g1[3] = (int)(((rows_remaining >> 16) & 0xFFFFu)           // tensor_dim1 hi
                | (64u << 16));                                // tile_dim0 = 64
    g1[4] = 64;                                                // tile_dim1 = 64
    g1[5] = 64;                                                // tensor_dim0_stride = 64
    g1[6] = 0;
    g1[7] = 0;

    v4i gz4  = {0, 0, 0, 0};                      // 2D tensor: groups 2/3 unused
    v8i gz8  = {0, 0, 0, 0, 0, 0, 0, 0};          // extra group unused
    __builtin_amdgcn_tensor_load_to_lds(g0, g1, gz4, gz4, gz8, 0);
}

__device__ __forceinline__ void tdm_wait() {
    __builtin_amdgcn_s_wait_tensorcnt(0);
}
#endif

// Synchronous staging fallback / Rinv staging (coalesced float4 loads).
__device__ __forceinline__ void stage_sync(const float* __restrict__ src,
                                           float* __restrict__ dstLds,
                                           int rowBase, int nrows, int tid) {
    const int r  = tid >> 3;          // 0..63
    const int c0 = (tid & 7) << 3;    // 0,8,...,56
    const int gr = rowBase + r;
    if (gr < nrows) {
        const float4 a = *(const float4*)(src + (size_t)gr * MCOLS + c0);
        const float4 b = *(const float4*)(src + (size_t)gr * MCOLS + c0 + 4);
        *(float4*)(&dstLds[r * LDSS + c0])     = a;
        *(float4*)(&dstLds[r * LDSS + c0 + 4]) = b;
    } else {
        const float4 z = {0.f, 0.f, 0.f, 0.f};
        *(float4*)(&dstLds[r * LDSS + c0])     = z;
        *(float4*)(&dstLds[r * LDSS + c0 + 4]) = z;
    }
}

// ---------------------------------------------------------------------------
// Kernel 1: G = X^T X   (64x64 fp32 accumulated via V_WMMA_F32_16X16X4_F32)
// block = 512 threads = 16 waves; each wave owns one 16x16 tile of G.
// Double-buffered LDS; wave 0 DMAs chunk i+1 while all waves compute chunk i.
// ---------------------------------------------------------------------------
__global__ __launch_bounds__(512) void gram_kernel(const float* __restrict__ X,
                                                   float* __restrict__ G,
                                                   int nrows) {
    __shared__ __align__(16) float sx[2][64 * LDSS];

    const int tid  = threadIdx.x;
    const int wave = tid >> 5;          // 0..15
    const int lane = tid & 31;
    const int ci   = (wave >> 2) << 4;  // A column-block base
    const int cj   = (wave & 3) << 4;   // B column-block base
    const int hk   = (lane >> 4) << 1;  // K offset within 4-step: 0 or 2
    const int lm   = lane & 15;         // M/N index within tile

    v8f acc = {0.f, 0.f, 0.f, 0.f, 0.f, 0.f, 0.f, 0.f};

    const int nchunks = (nrows + 63) >> 6;

#if USE_TDM
    const bool issuer = (tid < 32);     // wave 0 drives the TDM
    int buf = 0;
    {
        const int c0 = blockIdx.x;
        if (issuer && c0 < nchunks)
            tdm_load_tile(X + (size_t)(c0 << 6) * MCOLS,
                          (unsigned)(uintptr_t)&sx[0][0],
                          (unsigned)(nrows - (c0 << 6)));
    }
    for (int chunk = blockIdx.x; chunk < nchunks; chunk += gridDim.x) {
        if (issuer) tdm_wait();
        __syncthreads();                // buffer `buf` ready for all waves

        const int next = chunk + gridDim.x;
        if (issuer && next < nchunks)
            tdm_load_tile(X + (size_t)(next << 6) * MCOLS,
                          (unsigned)(uintptr_t)&sx[buf ^ 1][0],
                          (unsigned)(nrows - (next << 6)));

        const float* s = &sx[buf][0];
        #pragma unroll
        for (int kk = 0; kk < 64; kk += 4) {
            v2f a, b;
            a.x = s[(kk + hk)     * LDSS + ci + lm];
            a.y = s[(kk + hk + 1) * LDSS + ci + lm];
            b.x = s[(kk + hk)     * LDSS + cj + lm];
            b.y = s[(kk + hk + 1) * LDSS + cj + lm];
            acc = __builtin_amdgcn_wmma_f32_16x16x4_f32(
                false, a, false, b, (short)0, acc, false, false);
        }
        buf ^= 1;
    }
#else
    for (int chunk = blockIdx.x; chunk < nchunks; chunk += gridDim.x) {
        const int rowBase = chunk << 6;
        stage_sync(X, &sx[0][0], rowBase, nrows, tid);
        __syncthreads();
        const float* s = &sx[0][0];
        #pragma unroll
        for (int kk = 0; kk < 64; kk += 4) {
            v2f a, b;
            a.x = s[(kk + hk)     * LDSS + ci + lm];
            a.y = s[(kk + hk + 1) * LDSS + ci + lm];
            b.x = s[(kk + hk)     * LDSS + cj + lm];
            b.y = s[(kk + hk + 1) * LDSS + cj + lm];
            acc = __builtin_amdgcn_wmma_f32_16x16x4_f32(
                false, a, false, b, (short)0, acc, false, false);
        }
        __syncthreads();
    }
#endif

    // Reduce wave-private tile into G with fp32 global atomics.
    #pragma unroll
    for (int v = 0; v < 8; ++v) {
        const int m = v + ((lane >> 4) << 3);   // C/D layout: rows v / v+8
        atomicAdd(&G[(ci + m) * MCOLS + cj + lm], acc[v]);
    }
}

// ---------------------------------------------------------------------------
// Kernel 2: Cholesky G = R^T R, then Rinv = R^{-1} (upper triangular).
// One workgroup of 64 threads; everything in LDS.
// ---------------------------------------------------------------------------
__global__ __launch_bounds__(64) void chol_inv_kernel(const float* __restrict__ G,
                                                      float* __restrict__ Rinv) {
    __shared__ float s[64][65];
    __shared__ float ri[64][65];
    const int j = threadIdx.x;

    for (int i = 0; i < 64; ++i) s[i][j] = G[i * 64 + j];
    __syncthreads();

    // Right-looking Cholesky; upper triangle of s becomes R.
    for (int k = 0; k < 64; ++k) {
        if (j == k) s[k][k] = sqrtf(s[k][k]);
        __syncthreads();
        if (j > k) s[k][j] /= s[k][k];
        __syncthreads();
        for (int i = k + 1; i < 64; ++i)
            if (j >= i) s[i][j] -= s[k][i] * s[k][j];
        __syncthreads();
    }

    // Back-substitution: column j of Rinv solves R x = e_j.
    for (int i = 0; i < 64; ++i) ri[i][j] = 0.0f;
    ri[j][j] = 1.0f / s[j][j];
    for (int i = j - 1; i >= 0; --i) {
        float sum = 0.0f;
        for (int k = i + 1; k <= j; ++k) sum += s[i][k] * ri[k][j];
        ri[i][j] = -sum / s[i][i];
    }
    __syncthreads();

    for (int i = 0; i < 64; ++i) Rinv[i * 64 + j] = ri[i][j];
}

// ---------------------------------------------------------------------------
// Kernel 3: Q = X * Rinv   (500000x64 * 64x64 via V_WMMA_F32_16X16X4_F32)
// block = 512 threads = 16 waves; each wave owns one 16(row)x16(col) tile of
// a 64-row chunk. TDM double-buffered X staging; Rinv staged once per block.
// ---------------------------------------------------------------------------
__global__ __launch_bounds__(512) void apply_kernel(const float* __restrict__ X,
                                                    const float* __restrict__ Rinv,
                                                    float* __restrict__ Q,
                                                    int nrows) {
    __shared__ __align__(16) float sx[2][64 * LDSS];
    __shared__ __align__(16) float sr[64 * LDSS];

    const int tid  = threadIdx.x;
    const int wave = tid >> 5;
    const int lane = tid & 31;
    const int rt   = (wave >> 2) << 4;  // row-tile base within 64-row chunk
    const int cj   = (wave & 3) << 4;   // column-tile base
    const int hk   = (lane >> 4) << 1;  // 0 or 2
    const int lm   = lane & 15;

    // Stage Rinv once per block (small; plain coalesced loads).
    stage_sync(Rinv, &sr[0], 0, 64, tid);

    const int nchunks = (nrows + 63) >> 6;

#if USE_TDM
    const bool issuer = (tid < 32);
    int buf = 0;
    {
        const int c0 = blockIdx.x;
        if (issuer && c0 < nchunks)
            tdm_load_tile(X + (size_t)(c0 << 6) * MCOLS,
                          (unsigned)(uintptr_t)&sx[0][0],
                          (unsigned)(nrows - (c0 << 6)));
    }
    for (int chunk = blockIdx.x; chunk < nchunks; chunk += gridDim.x) {
        if (issuer) tdm_wait();
        __syncthreads();                // sx[buf] (and, 1st iter, sr) ready

        const int next = chunk + gridDim.x;
        if (issuer && next < nchunks)
            tdm_load_tile(X + (size_t)(next << 6) * MCOLS,
                          (unsigned)(uintptr_t)&sx[buf ^ 1][0],
                          (unsigned)(nrows - (next << 6)));

        const float* s = &sx[buf][0];
        v8f acc = {0.f, 0.f, 0.f, 0.f, 0.f, 0.f, 0.f, 0.f};
        #pragma unroll
        for (int kk = 0; kk < 64; kk += 4) {
            v2f a, b;
            a.x = s[(rt + lm) * LDSS + kk + hk];
            a.y = s[(rt + lm) * LDSS + kk + hk + 1];
            b.x = sr[(kk + hk)     * LDSS + cj + lm];
            b.y = sr[(kk + hk + 1) * LDSS + cj + lm];
            acc = __builtin_amdgcn_wmma_f32_16x16x4_f32(
                false, a, false, b, (short)0, acc, false, false);
        }

        const int rowBase = chunk << 6;
        #pragma unroll
        for (int v = 0; v < 8; ++v) {
            const int m   = v + ((lane >> 4) << 3);
            const int row = rowBase + rt + m;
            if (row < nrows) Q[(size_t)row * MCOLS + cj + lm] = acc[v];
        }
        buf ^= 1;
    }
#else
    __syncthreads();
    for (int chunk = blockIdx.x; chunk < nchunks; chunk += gridDim.x) {
        const int rowBase = chunk << 6;
        stage_sync(X, &sx[0][0], rowBase, nrows, tid);
        __syncthreads();

        const float* s = &sx[0][0];
        v8f acc = {0.f, 0.f, 0.f, 0.f, 0.f, 0.f, 0.f, 0.f};
        #pragma unroll
        for (int kk = 0; kk < 64; kk += 4) {
            v2f a, b;
            a.x = s[(rt + lm) * LDSS + kk + hk];
            a.y = s[(rt + lm) * LDSS + kk + hk + 1];
            b.x = sr[(kk + hk)     * LDSS + cj + lm];
            b.y = sr[(kk + hk + 1) * LDSS + cj + lm];
            acc = __builtin_amdgcn_wmma_f32_16x16x4_f32(
                false, a, false, b, (short)0, acc, false, false);
        }

        #pragma unroll
        for (int v = 0; v < 8; ++v) {
            const int m   = v + ((lane >> 4) << 3);
            const int row = rowBase + rt + m;
            if (row < nrows) Q[(size_t)row * MCOLS + cj + lm] = acc[v];
        }
        __syncthreads();
    }
#endif
}

extern "C" void kernel_launch(void* const* d_in, const int* in_sizes, int n_in,
                              void* d_out, int out_size, void* d_ws, size_t ws_size,
                              hipStream_t stream) {
    (void)n_in; (void)out_size; (void)ws_size;
    const float* X = (const float*)d_in[0];
    float* Q    = (float*)d_out;
    float* G    = (float*)d_ws;          // 64*64 floats
    float* Rinv = G + 64 * 64;           // 64*64 floats

    const int nrows = in_sizes[0] / MCOLS;   // 500000

    zero_kernel<<<16, 256, 0, stream>>>(G, 64 * 64);
    gram_kernel<<<256, 512, 0, stream>>>(X, G, nrows);
    chol_inv_kernel<<<1, 64, 0, stream>>>(G, Rinv);
    apply_kernel<<<512, 512, 0, stream>>>(X, Rinv, Q, nrows);
}